// SPINN_33328946217362
// MI455X (gfx1250) — compile-verified
//
#include <hip/hip_runtime.h>
#include <hip/hip_bf16.h>
#include <cstdint>
#include <cstddef>

// CDNA5 fp32 WMMA fragment types:
//  A (16x4 f32)  -> 2 VGPRs/lane, B (4x16 f32) -> 2 VGPRs/lane, C/D (16x16 f32) -> 8 VGPRs/lane
typedef __attribute__((ext_vector_type(2))) float v2f;
typedef __attribute__((ext_vector_type(4))) float v4f;
typedef __attribute__((ext_vector_type(8))) float v8f;

#define TILE_M 64
#define TILE_N 32
#define KC     64
#define LPAD   4        // keep LDS rows 16B-aligned AND kill 64-bank row-stride conflicts
#define KSPLIT 4        // deterministic K-split for the M=64 sequential-chain GEMM

// ---------------------------------------------------------------------------
// C[z][M][N] = A[M][k_begin:k_end] @ W[k_begin:k_end][N] (+ bias[N] if given)
// K-slice selected by blockIdx.z (slice z writes partial C + z*M*N).
// Block: 256 threads = 8 wave32 arranged 4(M) x 2(N) 16x16 tiles -> 64x32/block.
// float4 (global_load_b128 / ds_store_b128) staging, fp32 WMMA 16x16x4 compute.
// ---------------------------------------------------------------------------
__global__ __launch_bounds__(256)
void spinn_gemm_f32(const float* __restrict__ A, const float* __restrict__ W,
                    const float* __restrict__ bias, float* __restrict__ C,
                    int M, int N, int K, int kspan)
{
    __shared__ float lA[TILE_M][KC + LPAD];     // 64 x 68 floats (rows 272B, 16B-aligned)
    __shared__ float lW[KC][TILE_N + LPAD];     // 64 x 36 floats (rows 144B, 16B-aligned)

    const int tid  = threadIdx.x;
    const int wave = tid >> 5;                  // 0..7
    const int lane = tid & 31;
    const int half = lane >> 4;                 // 0: lanes 0-15, 1: lanes 16-31
    const int l15  = lane & 15;

    const int m_blk   = blockIdx.y * TILE_M;
    const int n_blk   = blockIdx.x * TILE_N;
    const int k_begin = blockIdx.z * kspan;
    const int k_end   = min(K, k_begin + kspan);
    const int wm = (wave & 3) * 16;             // wave's M offset in block tile
    const int wn = (wave >> 2) * 16;            // wave's N offset in block tile

    float* __restrict__ Cz = C + (size_t)blockIdx.z * (size_t)M * N;

    v8f acc = {};

    for (int k0 = k_begin; k0 < k_end; k0 += KC) {
        // Stage A block [64 x 64] as float4 (1024 vec4 over 256 threads)
        for (int i = tid; i < TILE_M * KC / 4; i += 256) {
            int r  = i >> 4;                    // / (KC/4)
            int c4 = (i & 15) << 2;
            int row = m_blk + r;
            v4f v = {};
            if (row < M) v = *(const v4f*)&A[(size_t)row * K + (k0 + c4)];
            *(v4f*)&lA[r][c4] = v;
        }
        // Stage W block [64 x 32] as float4 (512 vec4)
        for (int i = tid; i < KC * TILE_N / 4; i += 256) {
            int r  = i >> 3;                    // / (TILE_N/4)
            int c4 = (i & 7) << 2;
            *(v4f*)&lW[r][c4] = *(const v4f*)&W[(size_t)(k0 + r) * N + (n_blk + c4)];
        }
        __syncthreads();

        // 16 x v_wmma_f32_16x16x4_f32 per K-chunk
#pragma unroll
        for (int kk = 0; kk < KC; kk += 4) {
            v2f a, b;
            // A 16x4: lane holds row M=l15, K pair (kk + 2*half)
            a.x = lA[wm + l15][kk + 2 * half + 0];
            a.y = lA[wm + l15][kk + 2 * half + 1];
            // B 4x16: lane holds col N=l15, rows K = kk + 2*half (+0 in v0, +1 in v1)
            b.x = lW[kk + 2 * half + 0][wn + l15];
            b.y = lW[kk + 2 * half + 1][wn + l15];
            acc = __builtin_amdgcn_wmma_f32_16x16x4_f32(
                /*neg_a=*/false, a, /*neg_b=*/false, b,
                /*c_mod=*/(short)0, acc, /*reuse_a=*/false, /*reuse_b=*/false);
        }
        __syncthreads();
    }

    // D 16x16 f32 layout: VGPR r, lanes 0-15 -> M=r, lanes 16-31 -> M=r+8
    const int col = n_blk + wn + l15;
    const float bv = bias ? bias[col] : 0.0f;
#pragma unroll
    for (int r = 0; r < 8; ++r) {
        int row = m_blk + wm + 8 * half + r;
        if (row < M) Cz[(size_t)row * N + col] = acc[r] + bv;
    }
}

// ---------------------------------------------------------------------------
// Sequential shift/reduce control replay: one thread per batch element.
// ---------------------------------------------------------------------------
__global__ void spinn_control(const int* __restrict__ trans,
                              int* __restrict__ li, int* __restrict__ ri,
                              int* __restrict__ widx, int* __restrict__ bp_out,
                              int* __restrict__ isshift, int* __restrict__ fidx,
                              int B, int T, int L, int S)
{
    int b = blockIdx.x * blockDim.x + threadIdx.x;
    if (b >= B) return;
    int sp = 0, bp = 0;
    for (int t = 0; t < T; ++t) {
        int act  = trans[(size_t)b * T + t];   // [B, T] row-major
        int is_s = (act == 1);
        int is_r = (act == 2);
        ri[(size_t)t * B + b]      = min(max(sp - 1, 0), S - 1);
        li[(size_t)t * B + b]      = min(max(sp - 2, 0), S - 1);
        bp_out[(size_t)t * B + b]  = min(max(bp, 0), L - 1);
        widx[(size_t)t * B + b]    = is_s ? sp : (is_r ? sp - 2 : S);  // S/neg => drop
        isshift[(size_t)t * B + b] = is_s;
        sp += is_s ? 1 : (is_r ? -1 : 0);
        bp += is_s;
    }
    fidx[b] = min(max(sp - 1, 0), S - 1);
}

// ---------------------------------------------------------------------------
// x[b] = concat(sh[b][li[b]], sh[b][ri[b]])  (float4-vectorized gather)
// ---------------------------------------------------------------------------
__global__ void spinn_gather_x(const float* __restrict__ sh,
                               const int* __restrict__ li, const int* __restrict__ ri,
                               float* __restrict__ x, int B, int S, int H)
{
    int idx = blockIdx.x * blockDim.x + threadIdx.x;     // over B*2H/4
    int twoH4 = (2 * H) >> 2;
    if (idx >= B * twoH4) return;
    int b  = idx / twoH4;
    int j4 = (idx % twoH4) << 2;                         // 0..2H-4
    int slot = (j4 < H) ? li[b] : ri[b];
    int jj   = (j4 < H) ? j4 : j4 - H;
    *(v4f*)&x[(size_t)b * 2 * H + j4] =
        *(const v4f*)&sh[((size_t)b * S + slot) * H + jj];
}

// ---------------------------------------------------------------------------
// TreeLSTM gates (fixed-order sum over K-split partials + br), shift/reduce
// select, thin-stack scatter (mode='drop').
// ---------------------------------------------------------------------------
__device__ __forceinline__ float sigf(float x) { return 1.0f / (1.0f + __expf(-x)); }

__global__ void spinn_combine(const float* __restrict__ a,      // [ksplit][B][5H]
                              const float* __restrict__ br,
                              float* __restrict__ sh, float* __restrict__ sc,
                              const float* __restrict__ wordbuf, // [B*L][2H]
                              const int* __restrict__ li, const int* __restrict__ ri,
                              const int* __restrict__ widx, const int* __restrict__ bp,
                              const int* __restrict__ isshift,
                              int B, int S, int H, int L, int ksplit)
{
    int idx = blockIdx.x * blockDim.x + threadIdx.x;
    if (idx >= B * H) return;
    int b = idx / H, j = idx % H;
    int w = widx[b];
    if (w < 0 || w >= S) return;               // PAD / invalid reduce -> drop

    float wh, wc;
    if (isshift[b]) {
        const float* row = wordbuf + ((size_t)b * L + bp[b]) * (size_t)(2 * H);
        wh = row[j];
        wc = row[H + j];
    } else {
        // deterministic fixed-order partial sum: br + sum_z partial[z]
        float gi = br[j];
        float fl = br[H + j];
        float fr = br[2 * H + j];
        float go = br[3 * H + j];
        float gg = br[4 * H + j];
        const size_t pstride = (size_t)B * 5 * H;
        const size_t base    = (size_t)b * 5 * H;
        for (int z = 0; z < ksplit; ++z) {
            const float* ap = a + z * pstride + base;
            gi += ap[j];
            fl += ap[H + j];
            fr += ap[2 * H + j];
            go += ap[3 * H + j];
            gg += ap[4 * H + j];
        }
        float lc = sc[((size_t)b * S + li[b]) * H + j];
        float rc = sc[((size_t)b * S + ri[b]) * H + j];
        float c_new = sigf(fl) * lc + sigf(fr) * rc + sigf(gi) * tanhf(gg);
        wh = sigf(go) * tanhf(c_new);
        wc = c_new;
    }
    sh[((size_t)b * S + w) * H + j] = wh;
    sc[((size_t)b * S + w) * H + j] = wc;
}

__global__ void spinn_final(const float* __restrict__ sh, const int* __restrict__ fidx,
                            float* __restrict__ out, int B, int S, int H)
{
    int idx = blockIdx.x * blockDim.x + threadIdx.x;
    if (idx >= B * H) return;
    int b = idx / H, j = idx % H;
    out[idx] = sh[((size_t)b * S + fidx[b]) * H + j];
}

// ---------------------------------------------------------------------------
extern "C" void kernel_launch(void* const* d_in, const int* in_sizes, int n_in,
                              void* d_out, int out_size, void* d_ws, size_t ws_size,
                              hipStream_t stream)
{
    const float* sentence = (const float*)d_in[0];
    const int*   trans    = (const int*)d_in[1];
    const float* Ww       = (const float*)d_in[2];
    const float* bw       = (const float*)d_in[3];
    const float* Wr       = (const float*)d_in[4];
    const float* br       = (const float*)d_in[5];
    float* out = (float*)d_out;
    (void)n_in; (void)ws_size;

    const int twoH  = in_sizes[3];
    const int H     = twoH / 2;
    const int E     = in_sizes[2] / twoH;
    const int BL    = in_sizes[0] / E;          // B*L
    const int B     = out_size / H;
    const int L     = BL / B;
    const int T     = in_sizes[1] / B;
    const int S     = L + 1;
    const int fiveH = 5 * H;

    // K-split span for the reduce GEMM (multiple of KC, covers K in KSPLIT slices)
    const int kchunks = (twoH + KC - 1) / KC;
    const int kspan_r = ((kchunks + KSPLIT - 1) / KSPLIT) * KC;

    // Carve workspace (256B-aligned regions)
    char* p = (char*)d_ws;
    auto carve = [&](size_t bytes) {
        char* r = p;
        p += (bytes + 255) & ~(size_t)255;
        return r;
    };
    float* sh      = (float*)carve((size_t)B * S * H * sizeof(float));
    float* sc      = (float*)carve((size_t)B * S * H * sizeof(float));
    float* wordbuf = (float*)carve((size_t)BL * twoH * sizeof(float));
    float* xbuf    = (float*)carve((size_t)B * twoH * sizeof(float));
    float* abuf    = (float*)carve((size_t)KSPLIT * B * fiveH * sizeof(float));
    int*   c_li    = (int*)carve((size_t)T * B * sizeof(int));
    int*   c_ri    = (int*)carve((size_t)T * B * sizeof(int));
    int*   c_w     = (int*)carve((size_t)T * B * sizeof(int));
    int*   c_bp    = (int*)carve((size_t)T * B * sizeof(int));
    int*   c_s     = (int*)carve((size_t)T * B * sizeof(int));
    int*   c_f     = (int*)carve((size_t)B * sizeof(int));

    // Stacks must start zeroed every call (deterministic, capture-safe)
    hipMemsetAsync(sh, 0, (size_t)B * S * H * sizeof(float), stream);
    hipMemsetAsync(sc, 0, (size_t)B * S * H * sizeof(float), stream);

    // Replay transition control flow
    spinn_control<<<(B + 63) / 64, 64, 0, stream>>>(trans, c_li, c_ri, c_w, c_bp, c_s, c_f,
                                                    B, T, L, S);

    // Word projection: [BL x E] @ [E x 2H] + bw  (parallel WMMA GEMM, no K-split)
    {
        dim3 g(twoH / TILE_N, (BL + TILE_M - 1) / TILE_M, 1);
        spinn_gemm_f32<<<g, 256, 0, stream>>>(sentence, Ww, bw, wordbuf, BL, twoH, E, E);
    }

    // Sequential shift/reduce chain: gather -> K-split WMMA GEMM -> gates/scatter
    const int nelem_x4 = (B * twoH) / 4;
    const int nelem_h  = B * H;
    dim3 gemm_g(fiveH / TILE_N, (B + TILE_M - 1) / TILE_M, KSPLIT);
    for (int t = 0; t < T; ++t) {
        spinn_gather_x<<<(nelem_x4 + 255) / 256, 256, 0, stream>>>(
            sh, c_li + (size_t)t * B, c_ri + (size_t)t * B, xbuf, B, S, H);
        spinn_gemm_f32<<<gemm_g, 256, 0, stream>>>(
            xbuf, Wr, nullptr, abuf, B, fiveH, twoH, kspan_r);
        spinn_combine<<<(nelem_h + 255) / 256, 256, 0, stream>>>(
            abuf, br, sh, sc, wordbuf,
            c_li + (size_t)t * B, c_ri + (size_t)t * B, c_w + (size_t)t * B,
            c_bp + (size_t)t * B, c_s + (size_t)t * B, B, S, H, L, KSPLIT);
    }

    spinn_final<<<(nelem_h + 255) / 256, 256, 0, stream>>>(sh, c_f, out, B, S, H);
}